// CAN_12446815223813
// MI455X (gfx1250) — compile-verified
//
#include <hip/hip_runtime.h>
#include <math.h>

typedef __attribute__((ext_vector_type(2))) float v2f;
typedef __attribute__((ext_vector_type(8))) float v8f;

#define B_    8
#define NC_   32
#define NP_   8
#define ICII  256
#define IA_   16
#define DA_   16
#define DG    6
#define ID_   23      // 1 + 6 + 16
#define KD    2048    // NP_ * ICII
#define HP    24      // hat row pitch (padded to 96B): [0:6) geoms, [6:22) attrs
#define DC_   23
#define NI_   8

// ---------------------------------------------------------------------------
// Kernel A: transform. One wave per (b, nc, np). Via V_WMMA_F32_16X16X4_F32:
//   attrs[k,16] = x[...,7:23] (256x16) @ W2[nc,np] (16x16)     -> 4 WMMA / tile
//   geoms[k, 6] = [x[...,1:7],1] (256x7->8) @ W1[nc,np] (7x6)  -> 2 WMMA / tile
// ---------------------------------------------------------------------------
__global__ __launch_bounds__(256) void can_transform(
    const float* __restrict__ x, const float* __restrict__ W1,
    const float* __restrict__ W2, float* __restrict__ hat)
{
  const int lane = threadIdx.x & 31;
  const int wv   = threadIdx.x >> 5;
  const int gw   = blockIdx.x * 8 + wv;    // 0..2047
  const int b    = gw >> 8;
  const int rem  = gw & 255;
  const int nc   = rem >> 3;
  const int np   = rem & 7;
  const int n    = lane & 15;              // N (col) / M-row within tile
  const int half = lane >> 4;              // 0 or 1

  // B fragment layout: B[lane][v] holds K = 4*kk + (lane/16)*2 + v, N = lane%16
  const float* w2p = W2 + (size_t)((nc * NP_ + np) * IA_) * DA_;   // [K=a][N=da]
  v2f wb[4];
#pragma unroll
  for (int kk = 0; kk < 4; ++kk)
#pragma unroll
    for (int v = 0; v < 2; ++v)
      wb[kk][v] = w2p[(4 * kk + half * 2 + v) * DA_ + n];

  const float* w1p = W1 + (size_t)((nc * NP_ + np) * 7) * DG;      // [K=g][N=h]
  v2f wg[2];
#pragma unroll
  for (int kk = 0; kk < 2; ++kk)
#pragma unroll
    for (int v = 0; v < 2; ++v) {
      int K = 4 * kk + half * 2 + v;                // 0..7 (K=7 is zero pad)
      // branchless: clamp address, select value (avoids exec-masked loads)
      float wv1 = w1p[(K < 7 ? K : 0) * DG + (n < DG ? n : 0)];
      wg[kk][v] = (K < 7 && n < DG) ? wv1 : 0.0f;
    }

  float* dbase = hat + (size_t)((b * NC_ + nc) * KD + np * ICII) * HP;

  for (int mt = 0; mt < 16; ++mt) {
    // A fragment layout: A[lane][v] holds M = lane%16, K = 4*kk + (lane/16)*2 + v
    const float* xr = x + (size_t)((b * ICII + mt * 16 + n) * ID_);
    v2f aa[4], ag[2];
#pragma unroll
    for (int kk = 0; kk < 4; ++kk)
#pragma unroll
      for (int v = 0; v < 2; ++v)
        aa[kk][v] = xr[7 + 4 * kk + half * 2 + v];            // attrs cols 7..22
#pragma unroll
    for (int kk = 0; kk < 2; ++kk)
#pragma unroll
      for (int v = 0; v < 2; ++v) {
        int K = 4 * kk + half * 2 + v;
        float xv = xr[1 + (K < DG ? K : 0)];                  // branchless clamp
        ag[kk][v] = (K < DG) ? xv : ((K == DG) ? 1.0f : 0.0f);
      }

    v8f acc = {0.f, 0.f, 0.f, 0.f, 0.f, 0.f, 0.f, 0.f};
#pragma unroll
    for (int kk = 0; kk < 4; ++kk)
      acc = __builtin_amdgcn_wmma_f32_16x16x4_f32(
          false, aa[kk], false, wb[kk], (short)0, acc, false, false);

    v8f gac = {0.f, 0.f, 0.f, 0.f, 0.f, 0.f, 0.f, 0.f};
#pragma unroll
    for (int kk = 0; kk < 2; ++kk)
      gac = __builtin_amdgcn_wmma_f32_16x16x4_f32(
          false, ag[kk], false, wg[kk], (short)0, gac, false, false);

    // D layout: D[lane][v] holds N = lane%16, M = v + 8*(lane/16)
    float* drow = dbase + (size_t)(mt * 16) * HP;
#pragma unroll
    for (int v = 0; v < 8; ++v) {
      float* r = drow + (size_t)(v + 8 * half) * HP;
      r[DG + n] = acc[v];                 // attrs channel n
      if (n < DG) r[n] = gac[v];          // geom channel n (pad ch 22/23 unused)
    }
  }
}

// ---------------------------------------------------------------------------
// Kernel S: softmax stats. For each (b, icii): max and sum of exp over all
// (nc, np) logits (256 values). NI replication handled later as a x8 factor.
// ---------------------------------------------------------------------------
__global__ __launch_bounds__(256) void can_softmax_stats(
    const float* __restrict__ bl, float* __restrict__ smax,
    float* __restrict__ ssum)
{
  const int b = blockIdx.x;
  const int t = threadIdx.x;               // icii
  const float* p = bl + (size_t)b * NC_ * KD;

  float mx = -3.402823466e38f;
  for (int nc = 0; nc < NC_; ++nc)
#pragma unroll
    for (int np = 0; np < NP_; ++np)
      mx = fmaxf(mx, p[(size_t)nc * KD + np * ICII + t]);

  float s = 0.0f;
  for (int nc = 0; nc < NC_; ++nc)
#pragma unroll
    for (int np = 0; np < NP_; ++np)
      s += expf(p[(size_t)nc * KD + np * ICII + t] - mx);

  smax[b * ICII + t] = mx;
  ssum[b * ICII + t] = s;
}

__device__ __forceinline__ float waveReduceSum(float v) {
#pragma unroll
  for (int m = 16; m >= 1; m >>= 1) v += __shfl_xor(v, m, 32);
  return v;
}

__device__ __forceinline__ void loadRow24(const float* __restrict__ row,
                                          float* __restrict__ rr) {
  const float4* r4 = (const float4*)row;   // rows are 16B aligned (pitch 96B)
#pragma unroll
  for (int q = 0; q < 6; ++q) ((float4*)rr)[q] = r4[q];
}

// ---------------------------------------------------------------------------
// Kernel B: one routing iteration. One block per (b, nc); thread t owns
// icii = t, looping np = 0..7 (k = np*256 + t, all sharing the same stats).
// ---------------------------------------------------------------------------
__global__ __launch_bounds__(256) void can_route(
    const float* __restrict__ hat, float* __restrict__ bl,
    const float* __restrict__ smax, const float* __restrict__ ssum,
    float* __restrict__ out, int iter)
{
  const int b  = blockIdx.x >> 5;
  const int nc = blockIdx.x & 31;
  const int t  = threadIdx.x;
  const int lane = t & 31;
  const int wv   = t >> 5;

  const float* hrb = hat + (size_t)(b * NC_ + nc) * KD * HP;
  float* blp = bl + (size_t)(b * NC_ + nc) * KD;

  float mx = 0.0f, inv = 0.0f;
  if (iter > 0) {
    mx  = smax[b * ICII + t];
    inv = 1.0f / (8.0f * ssum[b * ICII + t]);   // x8 for the NI replicas
  }

  float csum = 0.0f;
  float sg[DG];  float wa[DA_];
#pragma unroll
  for (int d = 0; d < DG; ++d) sg[d] = 0.0f;
#pragma unroll
  for (int d = 0; d < DA_; ++d) wa[d] = 0.0f;

  for (int j = 0; j < NP_; ++j) {
    const int k = j * ICII + t;
    const float c = (iter == 0) ? (1.0f / 2048.0f) : expf(blp[k] - mx) * inv;
    float rr[HP];
    loadRow24(hrb + (size_t)k * HP, rr);
    csum += c;
#pragma unroll
    for (int d = 0; d < DG; ++d) sg[d] += c * rr[d];
#pragma unroll
    for (int d = 0; d < DA_; ++d) wa[d] += c * rr[DG + d];
  }

  // Cross-block reduction of 23 partials (wave32 shuffle + LDS)
  __shared__ float sPart[8][DC_];
  __shared__ float sFin[DC_];

  csum = waveReduceSum(csum);
#pragma unroll
  for (int d = 0; d < DG; ++d)  sg[d] = waveReduceSum(sg[d]);
#pragma unroll
  for (int d = 0; d < DA_; ++d) wa[d] = waveReduceSum(wa[d]);
  if (lane == 0) {
    sPart[wv][0] = csum;
#pragma unroll
    for (int d = 0; d < DG; ++d)  sPart[wv][1 + d] = sg[d];
#pragma unroll
    for (int d = 0; d < DA_; ++d) sPart[wv][7 + d] = wa[d];
  }
  __syncthreads();
  if (t < DC_) {
    float s = 0.0f;
#pragma unroll
    for (int w = 0; w < 8; ++w) s += sPart[w][t];
    sPart[0][t] = s;
  }
  __syncthreads();
  if (t == 0) {
    const float cs = sPart[0][0];
    const float invcs = 1.0f / cs;
    float s2 = 0.0f;
#pragma unroll
    for (int d = 0; d < DA_; ++d) { float w = sPart[0][7 + d]; s2 += w * w; }
    const float scale = s2 / ((1.0f + s2) * sqrtf(s2 + 1e-7f));
    sFin[0] = scale;
#pragma unroll
    for (int d = 0; d < DG; ++d)  sFin[1 + d] = sPart[0][1 + d] * invcs; // mean_geom
#pragma unroll
    for (int d = 0; d < DA_; ++d) sFin[7 + d] = scale * sPart[0][7 + d]; // scale*wa
  }
  __syncthreads();

  if (iter < 2) {
    float og[DG], oa[DA_];
#pragma unroll
    for (int d = 0; d < DG; ++d)  og[d] = sFin[1 + d];
#pragma unroll
    for (int d = 0; d < DA_; ++d) oa[d] = sFin[7 + d];
    for (int j = 0; j < NP_; ++j) {
      const int k = j * ICII + t;
      float rr[HP];
      loadRow24(hrb + (size_t)k * HP, rr);
      float agg = 0.0f, aga = 0.0f;
#pragma unroll
      for (int d = 0; d < DG; ++d)  agg += og[d] * rr[d];
#pragma unroll
      for (int d = 0; d < DA_; ++d) aga += oa[d] * rr[DG + d];
      const float ag = agg + 0.01f * aga;
      if (iter == 0) blp[k] = ag; else blp[k] += ag;
    }
  } else {
    // Final output: (B, NC, NI, 23), identical across NI
    if (t < NI_ * DC_) {
      const int ni = t / DC_, d = t % DC_;
      out[(size_t)((b * NC_ + nc) * NI_ + ni) * DC_ + d] = sFin[d];
    }
  }
}

extern "C" void kernel_launch(void* const* d_in, const int* in_sizes, int n_in,
                              void* d_out, int out_size, void* d_ws, size_t ws_size,
                              hipStream_t stream) {
  const float* x  = (const float*)d_in[0];   // (8,16,16,23)
  const float* W1 = (const float*)d_in[1];   // (32,8,7,6)
  const float* W2 = (const float*)d_in[2];   // (32,8,16,16)
  float* out = (float*)d_out;                // (8,32,8,23)

  float* ws   = (float*)d_ws;
  float* hat  = ws;                                        // 8*32*2048*24 floats
  float* bl   = hat + (size_t)B_ * NC_ * KD * HP;          // 8*32*2048 floats
  float* smax = bl + (size_t)B_ * NC_ * KD;                // 8*256
  float* ssum = smax + (size_t)B_ * ICII;                  // 8*256

  can_transform<<<256, 256, 0, stream>>>(x, W1, W2, hat);
  can_route<<<256, 256, 0, stream>>>(hat, bl, smax, ssum, out, 0);
  can_softmax_stats<<<8, 256, 0, stream>>>(bl, smax, ssum);
  can_route<<<256, 256, 0, stream>>>(hat, bl, smax, ssum, out, 1);
  can_softmax_stats<<<8, 256, 0, stream>>>(bl, smax, ssum);
  can_route<<<256, 256, 0, stream>>>(hat, bl, smax, ssum, out, 2);
}